// GATv2_46299747451297
// MI455X (gfx1250) — compile-verified
//
#include <hip/hip_runtime.h>
#include <hip/hip_bf16.h>
#include <math.h>

#define FDIM 128
#define HEADS 16
#define CPH 8
#define NEG_SLOPE 0.2f
#define ROWS_PER_BLK 64   // 4 row-tiles of 16 per block

typedef __attribute__((ext_vector_type(2))) float v2f;
typedef __attribute__((ext_vector_type(8))) float v8f;
typedef __attribute__((ext_vector_type(4))) unsigned int v4u;
typedef __attribute__((ext_vector_type(8))) int v8i;
typedef __attribute__((ext_vector_type(4))) int v4i;

__device__ __forceinline__ void atomic_max_f32(float* addr, float val) {
  // Sign-aware integer encoding; requires init to -INFINITY (0xFF800000).
  if (val >= 0.0f) atomicMax((int*)addr, __float_as_int(val));
  else             atomicMin((unsigned int*)addr, (unsigned int)__float_as_int(val));
}

__device__ __forceinline__ void atomic_add_f32(float* addr, float val) {
  // Native GLOBAL_ATOMIC_ADD_F32 (no CAS loop).
  unsafeAtomicAdd(addr, val);
}

// ---------------------------------------------------------------------------
// out[N,128] = in[N,128] @ W[128,128] for two weight matrices (blockIdx.z).
// W staged once per block into LDS by the Tensor Data Mover; each of 8 waves
// owns a 16-col block and computes 4 row-tiles (64 rows) via
// V_WMMA_F32_16X16X4_F32, reusing each B fragment across 4 WMMAs.
// ---------------------------------------------------------------------------
__global__ __launch_bounds__(256) void gemm128_wmma(
    const float* __restrict__ in,
    const float* __restrict__ W0, const float* __restrict__ W1,
    float* __restrict__ O0, float* __restrict__ O1, int nRows)
{
  __shared__ float Wlds[FDIM * FDIM];   // 64 KB of the 320 KB WGP LDS

  const float* W = blockIdx.z ? W1 : W0;
  float*       O = blockIdx.z ? O1 : O0;
  const int wave = threadIdx.x >> 5;
  const int lane = threadIdx.x & 31;
  const int hw   = lane >> 4;          // half-wave 0/1
  const int l16  = lane & 15;
  const int rowBase = blockIdx.x * ROWS_PER_BLK;
  const int c0      = wave * 16;

#if __has_builtin(__builtin_amdgcn_tensor_load_to_lds) && __has_builtin(__builtin_amdgcn_s_wait_tensorcnt)
  if (wave == 0) {
    // D# group0: count=1 | lds_addr=0 (only LDS alloc) | global_addr | type=2
    unsigned long long ga = (unsigned long long)(uintptr_t)W;
    v4u g0;
    g0.x = 1u;                                   // count=1, gather off
    g0.y = 0u;                                   // lds_addr = 0
    g0.z = (unsigned)(ga & 0xffffffffu);         // global_addr[31:0]
    g0.w = (unsigned)((ga >> 32) & 0x1ffffffu) | (2u << 30);  // addr[56:32] | type=2
    // D# group1: 1-D tensor of 16384 f32, tile = whole tensor
    v8i g1;
    g1[0] = 0x00020000;          // workgroup_mask=0, data_size=2 (4B)
    g1[1] = (int)(16384u << 16); // tensor_dim0[15:0] -> bits 63:48
    g1[2] = 0x00010000;          // tensor_dim0[31:16]=0, tensor_dim1=1
    g1[3] = (int)(16384u << 16); // tile_dim0 = 16384
    g1[4] = 0;                   // tile_dim1/2 unused
    g1[5] = 16384;               // tensor_dim0_stride lo
    g1[6] = 0;
    g1[7] = 0;
    v4i gz4 = {0, 0, 0, 0};
    v8i gz8 = {0, 0, 0, 0, 0, 0, 0, 0};
    __builtin_amdgcn_tensor_load_to_lds(g0, g1, gz4, gz4, gz8, 0);
    __builtin_amdgcn_s_wait_tensorcnt(0);
  }
  __syncthreads();
#else
  // Fallback: cooperative 256-thread copy (16 float4 each)
  for (int i = threadIdx.x; i < FDIM * FDIM / 4; i += 256)
    ((float4*)Wlds)[i] = ((const float4*)W)[i];
  __syncthreads();
#endif

  v8f acc0 = {}, acc1 = {}, acc2 = {}, acc3 = {};
  // A[m][k = 4*kk + 2*hw + v] per row-tile; clamp rows for ragged tail.
  int ra0 = rowBase + 0  + l16; if (ra0 > nRows - 1) ra0 = nRows - 1;
  int ra1 = rowBase + 16 + l16; if (ra1 > nRows - 1) ra1 = nRows - 1;
  int ra2 = rowBase + 32 + l16; if (ra2 > nRows - 1) ra2 = nRows - 1;
  int ra3 = rowBase + 48 + l16; if (ra3 > nRows - 1) ra3 = nRows - 1;
  const float* a0 = in + (size_t)ra0 * FDIM + 2 * hw;
  const float* a1 = in + (size_t)ra1 * FDIM + 2 * hw;
  const float* a2 = in + (size_t)ra2 * FDIM + 2 * hw;
  const float* a3 = in + (size_t)ra3 * FDIM + 2 * hw;
  const float* bl = Wlds + (size_t)(2 * hw) * FDIM + c0 + l16;

#pragma unroll
  for (int kk = 0; kk < 32; ++kk) {
    v2f b;                                   // B[k][n] from LDS, shared by 4 WMMAs
    b.x = bl[kk * 4 * FDIM];
    b.y = bl[kk * 4 * FDIM + FDIM];
    v2f fa0 = *(const v2f*)(a0 + kk * 4);
    v2f fa1 = *(const v2f*)(a1 + kk * 4);
    v2f fa2 = *(const v2f*)(a2 + kk * 4);
    v2f fa3 = *(const v2f*)(a3 + kk * 4);
    acc0 = __builtin_amdgcn_wmma_f32_16x16x4_f32(false, fa0, false, b, (short)0, acc0, false, false);
    acc1 = __builtin_amdgcn_wmma_f32_16x16x4_f32(false, fa1, false, b, (short)0, acc1, false, false);
    acc2 = __builtin_amdgcn_wmma_f32_16x16x4_f32(false, fa2, false, b, (short)0, acc2, false, false);
    acc3 = __builtin_amdgcn_wmma_f32_16x16x4_f32(false, fa3, false, b, (short)0, acc3, false, false);
  }

  // D[m = i + 8*hw][n = l16] per row-tile, predicated stores.
#pragma unroll
  for (int rt = 0; rt < 4; ++rt) {
    const v8f acc = rt == 0 ? acc0 : rt == 1 ? acc1 : rt == 2 ? acc2 : acc3;
    int r0 = rowBase + rt * 16;
    float* orow = O + (size_t)(r0 + 8 * hw) * FDIM + c0 + l16;
#pragma unroll
    for (int i = 0; i < 8; ++i)
      if (r0 + 8 * hw + i < nRows) orow[(size_t)i * FDIM] = acc[i];
  }
}

// ---------------------------------------------------------------------------
// Per-node init: out accum <- bias, segment-max <- -inf, denominator <- 0.
// ---------------------------------------------------------------------------
__global__ __launch_bounds__(256) void init_nodes(
    float* __restrict__ Cb, const float* __restrict__ bias,
    float* __restrict__ Mx, float* __restrict__ DN, int N)
{
  long long t = (long long)blockIdx.x * blockDim.x + threadIdx.x;
  if (t >= (long long)N * FDIM) return;
  int j = (int)(t & (FDIM - 1));
  Cb[t] = bias[j];
  if (j < HEADS) {
    size_t n = (size_t)(t >> 7);
    Mx[n * HEADS + j] = -INFINITY;
    DN[n * HEADS + j] = 0.0f;
  }
}

// ---------------------------------------------------------------------------
// Pass 1: logit[e,h] = sum_c att[h,c]*leakyrelu(xl[src,h,c]+xr[dst,h,c]);
// atomic segment-max into Mx[dst,h]. Thread per (edge,head): 16 consecutive
// threads cover one node row (512B) -> coalesced float4 gathers (L2-resident).
// Edges e >= E are the implicit self-loops (src=dst=e-E).
// ---------------------------------------------------------------------------
__global__ __launch_bounds__(256) void edge_logit(
    const float* __restrict__ XL, const float* __restrict__ XR,
    const float* __restrict__ att, const int* __restrict__ src,
    const int* __restrict__ dst, float* __restrict__ LG,
    float* __restrict__ Mx, long long E, int N)
{
  long long t = (long long)blockIdx.x * blockDim.x + threadIdx.x;
  long long total = (E + (long long)N) * HEADS;
  if (t >= total) return;
  long long e = t >> 4;
  int h = (int)(t & (HEADS - 1));
  int s, d;
  if (e < E) { s = src[e]; d = dst[e]; } else { s = d = (int)(e - E); }

  const float4* xl = (const float4*)(XL + (size_t)s * FDIM + h * CPH);
  const float4* xr = (const float4*)(XR + (size_t)d * FDIM + h * CPH);
  const float4* at = (const float4*)(att + h * CPH);
  float acc = 0.0f;
#pragma unroll
  for (int q = 0; q < 2; ++q) {
    float4 a = xl[q], b = xr[q], w = at[q];
    float e0 = a.x + b.x; e0 = e0 > 0.0f ? e0 : NEG_SLOPE * e0;
    float e1 = a.y + b.y; e1 = e1 > 0.0f ? e1 : NEG_SLOPE * e1;
    float e2 = a.z + b.z; e2 = e2 > 0.0f ? e2 : NEG_SLOPE * e2;
    float e3 = a.w + b.w; e3 = e3 > 0.0f ? e3 : NEG_SLOPE * e3;
    acc += e0 * w.x + e1 * w.y + e2 * w.z + e3 * w.w;
  }
  LG[t] = acc;
  atomic_max_f32(&Mx[(size_t)d * HEADS + h], acc);
}

// ---------------------------------------------------------------------------
// Pass 2: ex = exp(logit - max[dst]); store in place; atomic-add denominator.
// ---------------------------------------------------------------------------
__global__ __launch_bounds__(256) void edge_exp(
    float* __restrict__ LG, const float* __restrict__ Mx,
    float* __restrict__ DN, const int* __restrict__ dst, long long E, int N)
{
  long long t = (long long)blockIdx.x * blockDim.x + threadIdx.x;
  long long total = (E + (long long)N) * HEADS;
  if (t >= total) return;
  long long e = t >> 4;
  int h = (int)(t & (HEADS - 1));
  int d = (e < E) ? dst[e] : (int)(e - E);
  float ex = __expf(LG[t] - Mx[(size_t)d * HEADS + h]);
  LG[t] = ex;
  atomic_add_f32(&DN[(size_t)d * HEADS + h], ex);
}

// ---------------------------------------------------------------------------
// Pass 3: alpha = ex / (denom + eps); OUT[dst,h,:] += alpha * XL[src,h,:].
// 8 native GLOBAL_ATOMIC_ADD_F32 per (edge,head), resolved in L2.
// ---------------------------------------------------------------------------
__global__ __launch_bounds__(256) void edge_msg(
    const float* __restrict__ LG, const float* __restrict__ DN,
    const float* __restrict__ XL, const int* __restrict__ src,
    const int* __restrict__ dst, float* __restrict__ OUT, long long E, int N)
{
  long long t = (long long)blockIdx.x * blockDim.x + threadIdx.x;
  long long total = (E + (long long)N) * HEADS;
  if (t >= total) return;
  long long e = t >> 4;
  int h = (int)(t & (HEADS - 1));
  int s, d;
  if (e < E) { s = src[e]; d = dst[e]; } else { s = d = (int)(e - E); }

  float alpha = LG[t] / (DN[(size_t)d * HEADS + h] + 1e-16f);
  const float4* xl = (const float4*)(XL + (size_t)s * FDIM + h * CPH);
  float* o = OUT + (size_t)d * FDIM + h * CPH;
  float4 a = xl[0];
  atomic_add_f32(o + 0, a.x * alpha);
  atomic_add_f32(o + 1, a.y * alpha);
  atomic_add_f32(o + 2, a.z * alpha);
  atomic_add_f32(o + 3, a.w * alpha);
  float4 b = xl[1];
  atomic_add_f32(o + 4, b.x * alpha);
  atomic_add_f32(o + 5, b.y * alpha);
  atomic_add_f32(o + 6, b.z * alpha);
  atomic_add_f32(o + 7, b.w * alpha);
}

__global__ __launch_bounds__(256) void elu_inplace(float* __restrict__ p, long long n)
{
  long long t = (long long)blockIdx.x * blockDim.x + threadIdx.x;
  if (t >= n) return;
  float v = p[t];
  p[t] = v > 0.0f ? v : (__expf(v) - 1.0f);
}

// One wave32 per row of 128: float4 per lane, shuffle-reduce max & sum.
__global__ __launch_bounds__(256) void log_softmax_rows(float* __restrict__ O, int N)
{
  int row  = blockIdx.x * 8 + (threadIdx.x >> 5);
  int lane = threadIdx.x & 31;
  if (row >= N) return;
  float4* r = (float4*)(O + (size_t)row * FDIM);
  float4 v = r[lane];
  float m = fmaxf(fmaxf(v.x, v.y), fmaxf(v.z, v.w));
#pragma unroll
  for (int off = 16; off > 0; off >>= 1) m = fmaxf(m, __shfl_xor(m, off, 32));
  float s = __expf(v.x - m) + __expf(v.y - m) + __expf(v.z - m) + __expf(v.w - m);
#pragma unroll
  for (int off = 16; off > 0; off >>= 1) s += __shfl_xor(s, off, 32);
  float lse = m + __logf(s);
  v.x -= lse; v.y -= lse; v.z -= lse; v.w -= lse;
  r[lane] = v;
}

extern "C" void kernel_launch(void* const* d_in, const int* in_sizes, int n_in,
                              void* d_out, int out_size, void* d_ws, size_t ws_size,
                              hipStream_t stream)
{
  const float* x    = (const float*)d_in[0];
  const int*   ei   = (const int*)d_in[1];
  const float* Wl1  = (const float*)d_in[2];
  const float* Wr1  = (const float*)d_in[3];
  const float* att1 = (const float*)d_in[4];
  const float* b1   = (const float*)d_in[5];
  const float* Wl2  = (const float*)d_in[6];
  const float* Wr2  = (const float*)d_in[7];
  const float* att2 = (const float*)d_in[8];
  const float* b2   = (const float*)d_in[9];
  float* out = (float*)d_out;

  const int       N  = in_sizes[0] / FDIM;      // 50000
  const long long E  = in_sizes[1] / 2;         // 800000
  const int* src = ei;
  const int* dst = ei + E;
  const long long ET = E + (long long)N;        // + self-loops

  // Workspace carve-up (~138 MB): XL, XR, C1 [N*128]; LG [ET*16]; Mx,DN [N*16]
  float* XL = (float*)d_ws;
  float* XR = XL + (size_t)N * FDIM;
  float* C1 = XR + (size_t)N * FDIM;
  float* LG = C1 + (size_t)N * FDIM;
  float* Mx = LG + (size_t)ET * HEADS;
  float* DN = Mx + (size_t)N * HEADS;

  dim3 gemmGrid((unsigned)((N + ROWS_PER_BLK - 1) / ROWS_PER_BLK), 1, 2);
  long long eThreads = ET * HEADS;
  int eBlocks = (int)((eThreads + 255) / 256);
  int nBlocks = (int)(((long long)N * FDIM + 255) / 256);

  // -------- layer 1 --------
  gemm128_wmma<<<gemmGrid, 256, 0, stream>>>(x, Wl1, Wr1, XL, XR, N);
  init_nodes  <<<nBlocks, 256, 0, stream>>>(C1, b1, Mx, DN, N);
  edge_logit  <<<eBlocks, 256, 0, stream>>>(XL, XR, att1, src, dst, LG, Mx, E, N);
  edge_exp    <<<eBlocks, 256, 0, stream>>>(LG, Mx, DN, dst, E, N);
  edge_msg    <<<eBlocks, 256, 0, stream>>>(LG, DN, XL, src, dst, C1, E, N);
  elu_inplace <<<nBlocks, 256, 0, stream>>>(C1, (long long)N * FDIM);

  // -------- layer 2 (accumulate straight into d_out) --------
  gemm128_wmma<<<gemmGrid, 256, 0, stream>>>(C1, Wl2, Wr2, XL, XR, N);
  init_nodes  <<<nBlocks, 256, 0, stream>>>(out, b2, Mx, DN, N);
  edge_logit  <<<eBlocks, 256, 0, stream>>>(XL, XR, att2, src, dst, LG, Mx, E, N);
  edge_exp    <<<eBlocks, 256, 0, stream>>>(LG, Mx, DN, dst, E, N);
  edge_msg    <<<eBlocks, 256, 0, stream>>>(LG, DN, XL, src, dst, out, E, N);

  log_softmax_rows<<<(N + 7) / 8, 256, 0, stream>>>(out, N);
}